// LorentzBatchNormBase_55396488184452
// MI455X (gfx1250) — compile-verified
//
#include <hip/hip_runtime.h>
#include <math.h>

#define NROWS 1048576
#define NCH   64

typedef __attribute__((ext_vector_type(2))) float v2f;
typedef __attribute__((ext_vector_type(8))) float v8f;

constexpr int  B1 = 512;                    // col-sum blocks
constexpr int  B2 = 512;                    // alpha/var blocks
constexpr int  ROWS_PER_B1 = NROWS / B1;    // 2048
// workspace layout (float units)
constexpr long OFF_P1 = 0;                  // B1*64 col-sum partials
constexpr long OFF_MU = (long)B1 * NCH;     // 64: mu
constexpr long OFF_M  = OFF_MU + NCH;       // 64: m = (mu0, -mu1..)
constexpr long OFF_SC = OFF_M + NCH;        // 16: scalars (scale)
constexpr long OFF_P2 = OFF_SC + 16;        // B2: d^2 partials

// ---------------- Kernel 1: per-block column sums ----------------
__global__ __launch_bounds__(256) void colsum_kernel(const float* __restrict__ x,
                                                     float* __restrict__ p1) {
    const int t = threadIdx.x;
    const int c = t & 63;          // channel
    const int g = t >> 6;          // row group 0..3
    const long base = (long)blockIdx.x * ROWS_PER_B1;
    float s = 0.0f;
    for (int i = g; i < ROWS_PER_B1; i += 4)     // 1KB contiguous per iter per block
        s += x[(base + i) * NCH + c];
    __shared__ float lds[256];
    lds[t] = s;
    __syncthreads();
    if (t < 64)
        p1[(long)blockIdx.x * NCH + t] =
            lds[t] + lds[t + 64] + lds[t + 128] + lds[t + 192];
}

// ---------------- Kernel 2: finalize mu, m ----------------
__global__ void finalize_mu_kernel(float* __restrict__ ws) {
    const float* p1 = ws + OFF_P1;
    float* mu = ws + OFF_MU;
    float* m  = ws + OFF_M;
    const int c = threadIdx.x;     // 64 threads
    float s = 0.0f;
    for (int b = 0; b < B1; ++b) s += p1[(long)b * NCH + c];
    s *= (1.0f / (float)NROWS);
    __shared__ float lds[64];
    __shared__ float snS;
    lds[c] = s;
    __syncthreads();
    if (c == 0) {
        float q = lds[0] * lds[0];                 // -<s,s>_L = s0^2 - sum si^2
        for (int i = 1; i < 64; ++i) q -= lds[i] * lds[i];
        snS = sqrtf(fmaxf(q, 1e-9f));
    }
    __syncthreads();
    const float muc = s / snS;
    mu[c] = muc;
    m[c] = (c == 0) ? muc : -muc;                  // alpha = x . m
}

// ---------------- Kernel 3: alpha via WMMA (alpha = X @ m), d^2 partials ----------------
// A (16x4 f32) = m broadcast over rows: lanes 0-15 hold K={0,1}, lanes 16-31 K={2,3}.
// B (4x16 f32) = 16 x-rows as columns: lane j<16 holds (K=0,1,N=j); lane j+16 (K=2,3,N=j).
// D column j = alpha(row j) replicated -> lane j (and j+16) reads acc[0].
__global__ __launch_bounds__(256) void alpha_var_kernel(const float* __restrict__ x,
                                                        const float* __restrict__ ws,
                                                        float* __restrict__ p2) {
    const float* m = ws + OFF_M;
    const int t    = threadIdx.x;
    const int lane = t & 31;
    const int wave = t >> 5;                 // 8 waves / block
    const int row16 = lane & 15;
    const int koff  = (lane < 16) ? 0 : 2;

    // Preload broadcast-A fragments (16 K-steps x v2f)
    v2f a[16];
#pragma unroll
    for (int kb = 0; kb < 16; ++kb) {
        a[kb].x = m[kb * 4 + koff];
        a[kb].y = m[kb * 4 + koff + 1];
    }

    float acc_d2 = 0.0f;
    const int tilesTotal = NROWS / 16;                  // 65536
    const int waveGlobal = blockIdx.x * 8 + wave;
    const int nWaves     = B2 * 8;                      // 4096 -> 16 tiles each (uniform)
    for (int tile = waveGlobal; tile < tilesTotal; tile += nWaves) {
        const float* rowp = x + ((long)tile * 16 + row16) * NCH + koff;
        v8f acc = {0.f, 0.f, 0.f, 0.f, 0.f, 0.f, 0.f, 0.f};
#pragma unroll
        for (int kb = 0; kb < 16; ++kb) {
            v2f b = *(const v2f*)(rowp + kb * 4);       // global_load_b64, 8B aligned
            acc = __builtin_amdgcn_wmma_f32_16x16x4_f32(
                false, a[kb], false, b, (short)0, acc, false, false);
        }
        float alpha = fmaxf(acc[0], 1.0f + 1e-7f);
        float d = acoshf(alpha);
        if (lane < 16) acc_d2 += d * d;                 // lanes 16-31 are duplicates
    }

    __shared__ float red[256];
    red[t] = acc_d2;
    __syncthreads();
    for (int s = 128; s > 0; s >>= 1) {
        if (t < s) red[t] += red[t + s];
        __syncthreads();
    }
    if (t == 0) p2[blockIdx.x] = red[0];
}

// ---------------- Kernel 4: finalize var -> scale = gamma / sqrt(var+eps) ----------------
__global__ void finalize_var_kernel(float* __restrict__ ws, const float* __restrict__ gamma) {
    const float* p2 = ws + OFF_P2;
    float* sc = ws + OFF_SC;
    const int t = threadIdx.x;   // 256 threads
    __shared__ float red[256];
    red[t] = p2[t] + p2[t + 256];
    __syncthreads();
    for (int s = 128; s > 0; s >>= 1) {
        if (t < s) red[t] += red[t + s];
        __syncthreads();
    }
    if (t == 0) {
        float var = red[0] / (float)NROWS;
        sc[0] = gamma[0] / sqrtf(var + 1e-5f);
    }
}

// ---------------- Kernel 5: fused output pass (row per thread) ----------------
__global__ __launch_bounds__(256) void output_kernel(const float* __restrict__ x,
                                                     const float* __restrict__ beta,
                                                     const float* __restrict__ ws,
                                                     float* __restrict__ out) {
    __shared__ float sMu[64], sM[64], sBeta[64];
    __shared__ float sScale;
    const int t = threadIdx.x;
    if (t < 64) {
        sMu[t]   = ws[OFF_MU + t];
        sM[t]    = ws[OFF_M + t];
        sBeta[t] = beta[t];
    }
    if (t == 0) sScale = ws[OFF_SC];
    __syncthreads();

    const long row = (long)blockIdx.x * 256 + t;
    if (row >= NROWS) return;

    const float mu0   = sMu[0];
    const float beta0 = sBeta[0];
    const float scale = sScale;

    float vv[NCH];
    const float4* xr = (const float4*)(x + row * NCH);
#pragma unroll
    for (int i = 0; i < 16; ++i) {
        float4 q = xr[i];
        vv[4 * i + 0] = q.x; vv[4 * i + 1] = q.y;
        vv[4 * i + 2] = q.z; vv[4 * i + 3] = q.w;
    }

    // alpha = x . m  (Lorentz signs folded into m)
    float alpha = 0.0f;
#pragma unroll
    for (int c = 0; c < NCH; ++c) alpha = fmaf(vv[c], sM[c], alpha);
    alpha = fmaxf(alpha, 1.0f + 1e-7f);
    const float d = acoshf(alpha);
    const float w = d * rsqrtf(fmaxf(alpha * alpha - 1.0f, 1e-7f));

    // v = w * (x - alpha*mu)
#pragma unroll
    for (int c = 0; c < NCH; ++c) vv[c] = w * (vv[c] - alpha * sMu[c]);

    // v += (-v0)/(1+mu0) * (mu + e0)
    const float t1 = (-vv[0]) / (1.0f + mu0);
    vv[0] = fmaf(t1, sMu[0] + 1.0f, vv[0]);
#pragma unroll
    for (int c = 1; c < NCH; ++c) vv[c] = fmaf(t1, sMu[c], vv[c]);

    // v *= gamma/div
#pragma unroll
    for (int c = 0; c < NCH; ++c) vv[c] *= scale;

    // v += <beta,v>_L/(1+beta0) * (e0 + beta)
    float bv = -beta0 * vv[0];
#pragma unroll
    for (int c = 1; c < NCH; ++c) bv = fmaf(sBeta[c], vv[c], bv);
    const float t2 = bv / (1.0f + beta0);
    vv[0] = fmaf(t2, 1.0f + beta0, vv[0]);
#pragma unroll
    for (int c = 1; c < NCH; ++c) vv[c] = fmaf(t2, sBeta[c], vv[c]);

    // vn = sqrt(clamp(<v,v>_L, 1e-9)); out = cosh(vn)*beta + sinh(vn)*v/vn
    float q2 = -vv[0] * vv[0];
#pragma unroll
    for (int c = 1; c < NCH; ++c) q2 = fmaf(vv[c], vv[c], q2);
    const float vn = sqrtf(fmaxf(q2, 1e-9f));
    const float ch = coshf(vn);
    const float sh = sinhf(vn) / vn;

    float4* outr = (float4*)(out + row * NCH);
#pragma unroll
    for (int i = 0; i < 16; ++i) {
        float4 q;
        q.x = fmaf(ch, sBeta[4 * i + 0], sh * vv[4 * i + 0]);
        q.y = fmaf(ch, sBeta[4 * i + 1], sh * vv[4 * i + 1]);
        q.z = fmaf(ch, sBeta[4 * i + 2], sh * vv[4 * i + 2]);
        q.w = fmaf(ch, sBeta[4 * i + 3], sh * vv[4 * i + 3]);
        outr[i] = q;
    }
}

extern "C" void kernel_launch(void* const* d_in, const int* in_sizes, int n_in,
                              void* d_out, int out_size, void* d_ws, size_t ws_size,
                              hipStream_t stream) {
    const float* x     = (const float*)d_in[0];   // (N, 64)
    const float* gamma = (const float*)d_in[1];   // (1,)
    const float* beta  = (const float*)d_in[2];   // (64,)
    float* out = (float*)d_out;
    float* ws  = (float*)d_ws;

    colsum_kernel<<<B1, 256, 0, stream>>>(x, ws + OFF_P1);
    finalize_mu_kernel<<<1, 64, 0, stream>>>(ws);
    alpha_var_kernel<<<B2, 256, 0, stream>>>(x, ws, ws + OFF_P2);
    finalize_var_kernel<<<1, 256, 0, stream>>>(ws, gamma);
    output_kernel<<<NROWS / 256, 256, 0, stream>>>(x, beta, ws, out);
}